// SpikeFP64Exp_29875792511524
// MI455X (gfx1250) — compile-verified
//
#include <hip/hip_runtime.h>

// ---------------------------------------------------------------------------
// SpikeFP64Exp on MI455X (gfx1250): memory-bound decode of FP64 bit-rows.
// out row = zeros, then 1 at index (5-d), then mantissa bits m0..m(d-1),
// d = exponent-1023; all-zero unless d in [0,5]. Only floats x[1..16] used.
//
// Data movement strategy: per-lane ASYNC global->LDS copies (ASYNCcnt path,
// double-buffered, s_wait_asynccnt pipelining) touching only the first 68B of
// each 256B row; ds_load_b128 readback; NT b64 stores for the 24B/row output.
// ---------------------------------------------------------------------------

typedef __attribute__((ext_vector_type(4))) float f4;
typedef __attribute__((ext_vector_type(2))) float f2;
typedef __attribute__((ext_vector_type(4))) int   v4i;

// void-in-addrspace types (these casts compiled cleanly in round 2):
typedef __attribute__((address_space(1))) void gas_void;   // global
typedef __attribute__((address_space(3))) void las_void;   // LDS
// typed addrspace pointees matching the builtin signatures:
typedef __attribute__((address_space(1))) v4i  gas_v4i;
typedef __attribute__((address_space(3))) v4i  las_v4i;
typedef __attribute__((address_space(1))) int  gas_i32;
typedef __attribute__((address_space(3))) int  las_i32;

// --------------------- async global -> LDS primitives ----------------------
__device__ __forceinline__ void async_copy_b128(const float* g, float* l) {
  // (global int4* AS1, lds int4* AS3, imm offset, imm cpol)
  __builtin_amdgcn_global_load_async_to_lds_b128(
      (gas_v4i*)(gas_void*)const_cast<float*>(g),
      (las_v4i*)(las_void*)l, 0, 0);
}

__device__ __forceinline__ void async_copy_b32(const float* g, float* l) {
  __builtin_amdgcn_global_load_async_to_lds_b32(
      (gas_i32*)(gas_void*)const_cast<float*>(g),
      (las_i32*)(las_void*)l, 0, 0);
}

#if __has_builtin(__builtin_amdgcn_s_wait_asynccnt)
#define WAIT_ASYNC(n) __builtin_amdgcn_s_wait_asynccnt(n)
#else
#define WAIT_ASYNC(n) asm volatile("s_wait_asynccnt " #n ::: "memory")
#endif

// ---------------------------------------------------------------------------
#define BLOCK 256
#define SLOT_DW 20   // 80B per row slot: 16B aligned, 4x16B + 4B async chunks

// Issue the 5 async copies for one row (floats 0..16; float 0 is dead weight
// but keeps every transfer b128-aligned).
__device__ __forceinline__ void issue_row(const float* __restrict__ g,
                                          float* slot) {
  async_copy_b128(g + 0,  slot + 0);
  async_copy_b128(g + 4,  slot + 4);
  async_copy_b128(g + 8,  slot + 8);
  async_copy_b128(g + 12, slot + 12);
  async_copy_b32 (g + 16, slot + 16);
}

// Branchless decode of one staged row slot into 6 output floats.
__device__ __forceinline__ void decode_slot(const float* s, float o[6]) {
  f4 a  = *(const f4*)(s + 0);   // x[0..3]
  f4 b  = *(const f4*)(s + 4);   // x[4..7]
  f4 c  = *(const f4*)(s + 8);   // x[8..11]
  f4 mm = *(const f4*)(s + 12);  // m0..m3
  float m4 = s[16];              // m4

  int e = 0;                     // x[1] MSB ... x[11] LSB
  e |= (a.y > 0.5f) ? (1 << 10) : 0;
  e |= (a.z > 0.5f) ? (1 << 9)  : 0;
  e |= (a.w > 0.5f) ? (1 << 8)  : 0;
  e |= (b.x > 0.5f) ? (1 << 7)  : 0;
  e |= (b.y > 0.5f) ? (1 << 6)  : 0;
  e |= (b.z > 0.5f) ? (1 << 5)  : 0;
  e |= (b.w > 0.5f) ? (1 << 4)  : 0;
  e |= (c.x > 0.5f) ? (1 << 3)  : 0;
  e |= (c.y > 0.5f) ? (1 << 2)  : 0;
  e |= (c.z > 0.5f) ? (1 << 1)  : 0;
  e |= (c.w > 0.5f) ? 1         : 0;

  int  d     = e - 1023;
  bool valid = ((unsigned)d <= 5u);
  float m0 = mm.x, m1 = mm.y, m2 = mm.z, m3 = mm.w;

#pragma unroll
  for (int j = 0; j < 6; ++j) {
    int p   = j - 5 + d;         // position past the leading 1
    int idx = p - 1;             // mantissa index in [0,4] when used
    float mv = m0;
    mv = (idx == 1) ? m1 : mv;
    mv = (idx == 2) ? m2 : mv;
    mv = (idx == 3) ? m3 : mv;
    mv = (idx == 4) ? m4 : mv;
    float v = (valid && p >= 1) ? mv : 0.0f;
    v = (p == 0) ? 1.0f : v;     // p==0 implies 0<=d<=5
    o[j] = v;
  }
}

__global__ void __launch_bounds__(BLOCK)
SpikeFP64Exp_async(const float* __restrict__ x, float* __restrict__ out,
                   int n_rows) {
  __shared__ float stage[2 * BLOCK * SLOT_DW];   // 40KB double buffer

  const int tid    = blockIdx.x * BLOCK + (int)threadIdx.x;
  const int stride = (int)gridDim.x * BLOCK;

  float* slotA = &stage[(int)threadIdx.x * SLOT_DW];
  float* slotB = slotA + BLOCK * SLOT_DW;

  // Prologue: stage buffer A for the first iteration.
  if (tid < n_rows) issue_row(x + (size_t)tid * 64, slotA);

  int buf = 0;
  for (int i = tid; i < n_rows; i += stride) {
    float* cur = buf ? slotB : slotA;
    float* nxt = buf ? slotA : slotB;

    // Deep prefetch: pull row (i + 2*stride) toward L2 while the async stage
    // covers (i + stride).
    long pf = (long)i + 2L * stride;
    if (pf < (long)n_rows) {
      const float* q = x + pf * 64;
      __builtin_prefetch(q + 0,  0, 1);
      __builtin_prefetch(q + 16, 0, 1);
    }

    int nx = i + stride;
    if (nx < n_rows) {
      issue_row(x + (size_t)nx * 64, nxt);
      // 5 newly-issued ops outstanding allowed => current buffer's 5 retired
      // (async loads complete in order).
      WAIT_ASYNC(5);
    } else {
      WAIT_ASYNC(0);
    }
    asm volatile("" ::: "memory");   // keep LDS reads below the wait

    float o[6];
    decode_slot(cur, o);

    // 24B/row output, contiguous across lanes: three NT b64 stores.
    float* ob = out + (size_t)i * 6;
    f2 s0 = {o[0], o[1]};
    f2 s1 = {o[2], o[3]};
    f2 s2 = {o[4], o[5]};
    __builtin_nontemporal_store(s0, (f2*)(ob + 0));
    __builtin_nontemporal_store(s1, (f2*)(ob + 2));
    __builtin_nontemporal_store(s2, (f2*)(ob + 4));

    buf ^= 1;
  }
}

extern "C" void kernel_launch(void* const* d_in, const int* in_sizes, int n_in,
                              void* d_out, int out_size, void* d_ws,
                              size_t ws_size, hipStream_t stream) {
  const float* x = (const float*)d_in[0];
  float* out = (float*)d_out;

  const long total = (long)in_sizes[0];
  const int n_rows = (int)(total / 64);

  // Grid-stride: 1024 blocks x 256 threads (8 wave32/block) -> ~262k lanes,
  // ~8 pipelined iterations over 2M rows; plenty of waves to saturate HBM.
  const int block = BLOCK;
  int grid = 1024;
  long need = ((long)n_rows + block - 1) / block;
  if (need < grid) grid = (int)(need > 0 ? need : 1);

  SpikeFP64Exp_async<<<grid, block, 0, stream>>>(x, out, n_rows);
}